// PostProcess_13752485282104
// MI455X (gfx1250) — compile-verified
//
#include <hip/hip_runtime.h>
#include <hip/hip_bf16.h>
#include <stdint.h>

// ---------------------------------------------------------------------------
// DETR PostProcess for MI455X (gfx1250).
// Shapes fixed by the reference: bs=32, Q=4*1000=4000, C=256.
// Per batch: exact top-4000 (sorted desc) of 1,024,000 sigmoid(logits).
// sigmoid is monotone -> select on raw logits via a sortable u32 key,
// apply sigmoid only to the 4000 winners.
//
// One wave32-sized workgroup (512 threads = 16 waves) per batch.
//  Pass 1-3 : radix select (10/11/11 bits) w/ LDS histograms -> exact kth key
//  Pass 4   : gather exactly K winners into LDS via ds_add_rtn counters
//  Sort     : 4096-wide bitonic sort in LDS (descending)
//  Emit     : scores = sigmoid(logit), labels = 1, boxes cxcywh->xyxy * (w,h,w,h)
// All 4 streaming passes stage 8KB tiles into LDS with
// global_load_async_to_lds_b128 (double buffered, ASYNCcnt). After pass 1 the
// 128MB of logits is resident in the 192MB L2, so passes 2-4 run at L2 BW;
// HBM traffic ~= one read of the inputs (~5.6us floor at 23.3 TB/s).
// ---------------------------------------------------------------------------

#define BS        32
#define NQ        4000
#define NC        256
#define NELEM     (NQ * NC)          // 1,024,000 logits per batch
#define KSEL      4000
#define KPAD      4096
#define NTHREADS  512
#define NVEC      (NELEM / 4)        // 256,000 float4 per batch
#define NTILES    (NVEC / NTHREADS)  // 500 tiles, exact

#define USE_ASYNC 1

// float -> order-preserving unsigned key (bigger float == bigger key)
__device__ __forceinline__ unsigned keyOf(float f) {
  unsigned u = __float_as_uint(f);
  return u ^ ((u >> 31) ? 0xFFFFFFFFu : 0x80000000u);
}
__device__ __forceinline__ float invKey(unsigned k) {
  unsigned u = (k & 0x80000000u) ? (k ^ 0x80000000u) : ~k;
  return __uint_as_float(u);
}

// Issue one 16B/lane async copy global -> LDS (tile = NTHREADS float4 = 8KB).
// Each lane later reads back only its own slot, so completion is tracked by
// the lane's own per-wave ASYNCcnt; no workgroup barrier needed in the loop.
__device__ __forceinline__ void tile_issue(const float4* gsrc, int tile,
                                           float4* buf, int tid) {
#if USE_ASYNC
  unsigned lds_off = (unsigned)(uintptr_t)&buf[tid];  // low 32 bits of generic
                                                      // LDS ptr == wave-relative
                                                      // LDS byte address
  unsigned long long ga =
      (unsigned long long)(uintptr_t)&gsrc[(size_t)tile * NTHREADS + tid];
  asm volatile("global_load_async_to_lds_b128 %0, %1, off"
               :: "v"(lds_off), "v"(ga)
               : "memory");
#else
  buf[tid] = gsrc[(size_t)tile * NTHREADS + tid];
#endif
}
__device__ __forceinline__ void tile_wait(int more_in_flight) {
#if USE_ASYNC
  if (more_in_flight) asm volatile("s_wait_asynccnt 0x1" ::: "memory");
  else                asm volatile("s_wait_asynccnt 0x0" ::: "memory");
#endif
}

__global__ __launch_bounds__(NTHREADS)
void detr_postprocess(const float* __restrict__ pred_logits,
                      const float* __restrict__ pred_boxes,
                      const int*   __restrict__ target_sizes,
                      float*       __restrict__ out) {
  __shared__ unsigned hist[2048];           // radix histogram / suffix sums
  __shared__ unsigned sKey[KPAD];           // selected keys (sort payload)
  __shared__ unsigned sIdx[KPAD];           // selected flat indices in [0,Q*C)
  __shared__ float4   stage[2][NTHREADS];   // async double buffer (16KB)
  __shared__ unsigned sBin, sAbove, sCtrHigh, sCtrTie;

  const int tid = threadIdx.x;
  const int b   = blockIdx.x;
  const float4* gsrc = (const float4*)(pred_logits + (size_t)b * NELEM);

  // ---------------- radix select: find exact kth-largest key ----------------
  unsigned countAbove = 0;   // elements strictly above current selected prefix
  unsigned prefix     = 0;   // selected key bits so far (grows 10 -> 21 -> 32)

  for (int pass = 0; pass < 3; ++pass) {
    const int nb = (pass == 0) ? 1024 : 2048;
    for (int i = tid; i < nb; i += NTHREADS) hist[i] = 0u;
    __syncthreads();

    // streaming histogram with async LDS staging
    tile_issue(gsrc, 0, stage[0], tid);
    for (int t = 0; t < NTILES; ++t) {
      if (t + 1 < NTILES) tile_issue(gsrc, t + 1, stage[(t + 1) & 1], tid);
      tile_wait(t + 1 < NTILES);
      float4 v = stage[t & 1][tid];
      const float elem[4] = {v.x, v.y, v.z, v.w};
      #pragma unroll
      for (int j = 0; j < 4; ++j) {
        unsigned key = keyOf(elem[j]);
        if (pass == 0) {
          atomicAdd(&hist[key >> 22], 1u);
        } else if (pass == 1) {
          if ((key >> 22) == prefix) atomicAdd(&hist[(key >> 11) & 2047u], 1u);
        } else {
          if ((key >> 11) == prefix) atomicAdd(&hist[key & 2047u], 1u);
        }
      }
    }
    __syncthreads();

    // in-place parallel suffix sum over hist[0..nb): hist[i] = #elems in bins >= i
    for (int off = 1; off < nb; off <<= 1) {
      unsigned tmp[4]; int c = 0;
      for (int i = tid; i < nb; i += NTHREADS)
        tmp[c++] = hist[i] + ((i + off < nb) ? hist[i + off] : 0u);
      __syncthreads();
      c = 0;
      for (int i = tid; i < nb; i += NTHREADS) hist[i] = tmp[c++];
      __syncthreads();
    }

    // unique crossing bin: suffix[i] >= need > suffix[i+1]
    const unsigned need = (unsigned)KSEL - countAbove;
    for (int i = tid; i < nb; i += NTHREADS) {
      unsigned s  = hist[i];
      unsigned sn = (i + 1 < nb) ? hist[i + 1] : 0u;
      if (s >= need && sn < need) { sBin = (unsigned)i; sAbove = sn; }
    }
    __syncthreads();
    const unsigned bsel = sBin;
    countAbove += sAbove;
    prefix = (pass == 0) ? bsel : ((prefix << 11) | bsel);
    __syncthreads();
  }
  const unsigned kthKey     = prefix;                    // exact kth-largest key
  const unsigned tiesNeeded = (unsigned)KSEL - countAbove; // >= 1

  // ---------------- gather exactly K winners into LDS -----------------------
  if (tid == 0) { sCtrHigh = 0u; sCtrTie = 0u; }
  __syncthreads();

  tile_issue(gsrc, 0, stage[0], tid);
  for (int t = 0; t < NTILES; ++t) {
    if (t + 1 < NTILES) tile_issue(gsrc, t + 1, stage[(t + 1) & 1], tid);
    tile_wait(t + 1 < NTILES);
    float4 v = stage[t & 1][tid];
    const float elem[4] = {v.x, v.y, v.z, v.w};
    const unsigned base = ((unsigned)t * NTHREADS + (unsigned)tid) * 4u;
    #pragma unroll
    for (int j = 0; j < 4; ++j) {
      unsigned key = keyOf(elem[j]);
      if (key > kthKey) {
        unsigned p = atomicAdd(&sCtrHigh, 1u);          // p < countAbove
        sKey[p] = key; sIdx[p] = base + (unsigned)j;
      } else if (key == kthKey) {
        unsigned p = atomicAdd(&sCtrTie, 1u);
        if (p < tiesNeeded) {
          sKey[countAbove + p] = key; sIdx[countAbove + p] = base + (unsigned)j;
        }
      }
    }
  }
  __syncthreads();
  for (int i = KSEL + tid; i < KPAD; i += NTHREADS) { sKey[i] = 0u; sIdx[i] = 0u; }
  __syncthreads();

  // ---------------- bitonic sort (descending by key) in LDS -----------------
  for (unsigned size = 2; size <= KPAD; size <<= 1) {
    for (unsigned stride = size >> 1; stride > 0; stride >>= 1) {
      for (unsigned t = tid; t < (KPAD / 2); t += NTHREADS) {
        unsigned i = 2u * t - (t & (stride - 1u));
        unsigned j = i + stride;
        bool desc = ((i & size) == 0u);
        unsigned ki = sKey[i], kj = sKey[j];
        bool sw = desc ? (ki < kj) : (ki > kj);
        if (sw) {
          sKey[i] = kj; sKey[j] = ki;
          unsigned ti = sIdx[i]; sIdx[i] = sIdx[j]; sIdx[j] = ti;
        }
      }
      __syncthreads();
    }
  }

  // ---------------- emit scores / labels / scaled boxes ---------------------
  const float img_h = (float)target_sizes[2 * b];
  const float img_w = (float)target_sizes[2 * b + 1];
  float* scores = out;
  float* labels = out + (size_t)BS * KSEL;
  float* boxesO = out + (size_t)2 * BS * KSEL;
  const float4* boxIn = (const float4*)pred_boxes;   // [bs, Q, 4] cxcywh

  for (int r = tid; r < KSEL; r += NTHREADS) {
    unsigned key = sKey[r];
    unsigned idx = sIdx[r];
    float val = invKey(key);
    scores[(size_t)b * KSEL + r] = 1.0f / (1.0f + expf(-val));
    labels[(size_t)b * KSEL + r] = 1.0f;              // module forces labels=1

    unsigned q = idx / NC;                             // topk_idx // C
    float4 bx = boxIn[(size_t)b * NQ + q];
    float x0 = bx.x - 0.5f * bx.z, y0 = bx.y - 0.5f * bx.w;
    float x1 = bx.x + 0.5f * bx.z, y1 = bx.y + 0.5f * bx.w;
    float4 o = make_float4(x0 * img_w, y0 * img_h, x1 * img_w, y1 * img_h);
    ((float4*)boxesO)[(size_t)b * KSEL + r] = o;
  }
}

extern "C" void kernel_launch(void* const* d_in, const int* in_sizes, int n_in,
                              void* d_out, int out_size, void* d_ws, size_t ws_size,
                              hipStream_t stream) {
  (void)in_sizes; (void)n_in; (void)out_size; (void)d_ws; (void)ws_size;
  const float* pred_logits  = (const float*)d_in[0];  // [32, 4, 1000, 256] f32
  const float* pred_boxes   = (const float*)d_in[1];  // [32, 4, 1000, 4]   f32
  const int*   target_sizes = (const int*)d_in[2];    // [32, 2] i32 (h, w)
  detr_postprocess<<<BS, NTHREADS, 0, stream>>>(pred_logits, pred_boxes,
                                                target_sizes, (float*)d_out);
}